// PSimpleMeanAggregator_50491635532346
// MI455X (gfx1250) — compile-verified
//
#include <hip/hip_runtime.h>

typedef __attribute__((ext_vector_type(2))) float v2f;
typedef __attribute__((ext_vector_type(4))) float v4f;
typedef __attribute__((ext_vector_type(8))) float v8f;

#define K_NEIB 32
#define D 128
#define TILE_M 16
#define LDS_STRIDE 132   // 132 mod 64 = 4 -> conflict-free row reads in WMMA phase

__global__ __launch_bounds__(256) void graphsage_fused_kernel(
    const float* __restrict__ neib_feats,   // [N*K, D]
    const int*   __restrict__ neib_ids,     // [N, K]
    const float* __restrict__ emb,          // [N_NODES+1]
    const float* __restrict__ W,            // [D, D] row-major
    const float* __restrict__ bias,         // [D]
    float*       __restrict__ out,          // [N, D]
    int n)
{
    __shared__ float ldsAgg[TILE_M * LDS_STRIDE];

    const int lane = threadIdx.x & 31;
    const int wv   = threadIdx.x >> 5;          // wave id 0..7
    const int blockBase = blockIdx.x * TILE_M;

    // ---------------- Phase A: gather + softmax + weighted aggregation ----
    // Each wave handles tile rows {wv, wv+8}. Lane k owns neighbor k for the
    // softmax; for the feature accumulation each lane owns a float4 slice of D.
    #pragma unroll
    for (int i = 0; i < 2; ++i) {
        const int r    = wv + i * 8;            // row in tile, 0..15
        const int node = blockBase + r;
        v4f acc = {0.f, 0.f, 0.f, 0.f};
        if (node < n) {
            // gather per-neighbor scalar weight
            const int   nid = neib_ids[node * K_NEIB + lane];
            const float e   = emb[nid];
            // wave32 softmax over K=32 neighbors
            float m = e;
            #pragma unroll
            for (int off = 16; off >= 1; off >>= 1)
                m = fmaxf(m, __shfl_xor(m, off));
            const float ex = __expf(e - m);
            float s = ex;
            #pragma unroll
            for (int off = 16; off >= 1; off >>= 1)
                s += __shfl_xor(s, off);
            const float wlane = ex / s;         // lane k holds softmax weight k

            // stream K rows of 128 floats (512 B / row across the wave),
            // non-temporal: 819 MB single-use stream, keep W/emb in L2
            const v4f* base =
                (const v4f*)(neib_feats + (size_t)node * (K_NEIB * D));
            #pragma unroll
            for (int k = 0; k < K_NEIB; ++k) {
                const float wk = __shfl(wlane, k);   // readlane broadcast
                v4f f = __builtin_nontemporal_load(&base[k * (D / 4) + lane]);
                acc.x = fmaf(wk, f.x, acc.x);
                acc.y = fmaf(wk, f.y, acc.y);
                acc.z = fmaf(wk, f.z, acc.z);
                acc.w = fmaf(wk, f.w, acc.w);
            }
        }
        *(v4f*)&ldsAgg[r * LDS_STRIDE + lane * 4] = acc;
    }
    __syncthreads();

    // ---------------- Phase B: out = agg @ W + b via V_WMMA_F32_16X16X4_F32
    // Wave wv computes the 16x16 output tile covering columns [wv*16, wv*16+16).
    // A-fragment layout (16x4 f32): lanes 0-15 hold K=0,1; lanes 16-31 K=2,3.
    const int col  = wv * 16 + (lane & 15);
    const int half = lane >> 4;                 // 0: K+{0,1} / M 0-7, 1: K+{2,3} / M 8-15
    const int arow = lane & 15;
    v8f c = {};
    #pragma unroll
    for (int k4 = 0; k4 < D / 4; ++k4) {        // 32 k-steps of K=4
        const int kk = k4 * 4 + half * 2;
        v2f a, bm;
        a.x  = ldsAgg[arow * LDS_STRIDE + kk];
        a.y  = ldsAgg[arow * LDS_STRIDE + kk + 1];
        bm.x = W[kk * D + col];
        bm.y = W[(kk + 1) * D + col];
        c = __builtin_amdgcn_wmma_f32_16x16x4_f32(
                false, a, false, bm, (short)0, c, false, false);
    }

    // C/D layout: VGPR r -> M = r + 8*half, N = lane&15
    const float bcol    = bias[col];
    const int   rowbase = blockBase + half * 8;
    #pragma unroll
    for (int r = 0; r < 8; ++r) {
        const int row = rowbase + r;
        if (row < n)
            __builtin_nontemporal_store(c[r] + bcol, &out[(size_t)row * D + col]);
    }
}

extern "C" void kernel_launch(void* const* d_in, const int* in_sizes, int n_in,
                              void* d_out, int out_size, void* d_ws, size_t ws_size,
                              hipStream_t stream) {
    // setup_inputs order:
    // 0: node_feats [N,128] f32 (unused by reference)
    // 1: neib_feats [N*K,128] f32
    // 2: node_ids   [N] i32 (unused by reference)
    // 3: neib_ids   [N,K] i32
    // 4: emb        [N_NODES+1,1] f32
    // 5: W          [128,128] f32
    // 6: b          [128] f32
    const float* neib_feats = (const float*)d_in[1];
    const int*   neib_ids   = (const int*)d_in[3];
    const float* emb        = (const float*)d_in[4];
    const float* W          = (const float*)d_in[5];
    const float* b          = (const float*)d_in[6];
    float* out = (float*)d_out;

    const int n = in_sizes[0] / D;              // N from node_feats size
    const int blocks = (n + TILE_M - 1) / TILE_M;

    hipLaunchKernelGGL(graphsage_fused_kernel, dim3(blocks), dim3(256), 0, stream,
                       neib_feats, neib_ids, emb, W, b, out, n);
    (void)d_ws; (void)ws_size; (void)n_in; (void)out_size;
}